// atomic_predictor_49950469652660
// MI455X (gfx1250) — compile-verified
//
#include <hip/hip_runtime.h>

typedef float v2f __attribute__((ext_vector_type(2)));
typedef float v8f __attribute__((ext_vector_type(8)));

#define D_IN 128
#define NH   16

// ---------------------------------------------------------------- prologue
__global__ void zero_out_kernel(float* __restrict__ out, int n) {
  int i = blockIdx.x * blockDim.x + threadIdx.x;
  if (i < n) out[i] = 0.0f;
}

// W01[d][j] = sum_k W0[d][k] * W1[k][j]   (128x16, fp32, 2048 elements)
__global__ void fuse_w01_kernel(const float* __restrict__ W0,
                                const float* __restrict__ W1,
                                float* __restrict__ W01) {
  int i = blockIdx.x * blockDim.x + threadIdx.x;
  if (i >= D_IN * NH) return;
  int d = i >> 4, j = i & 15;
  float acc = 0.0f;
#pragma unroll
  for (int k = 0; k < NH; ++k) acc += W0[d * NH + k] * W1[k * NH + j];
  W01[i] = acc;
}

// ---------------------------------------------------------------- main kernel
// One wave32 per 16-atom tile (grid-stride). h = x_tile[16x128] @ W01[128x16]
// via 32 chained V_WMMA_F32_16X16X4_F32; pred = h @ W2 via shfl reductions;
// segment-sum with a one-atomic-per-tile fast path (structure ids are sorted).
__global__ __launch_bounds__(256) void mlp_segsum_kernel(
    const float* __restrict__ x, const float* __restrict__ W01,
    const float* __restrict__ W2, const int* __restrict__ structure,
    float* __restrict__ out, int n_atoms, int n_tiles) {
  const int lane = threadIdx.x & 31;
  const int wave = threadIdx.x >> 5;
  const int wave_global = blockIdx.x * (blockDim.x >> 5) + wave;
  const int wave_stride = gridDim.x * (blockDim.x >> 5);

  const int col = lane & 15;      // column / atom-row selector
  const int hi  = lane >> 4;      // 0: lanes 0-15, 1: lanes 16-31

  // B operand (W01) in 32-bit 4x16 B layout, resident in VGPRs for all tiles:
  // VGPR0: lanes0-15 K=r0, lanes16-31 K=r0+2 ; VGPR1: K=r0+1 / K=r0+3
  v2f bb[32];
#pragma unroll
  for (int kk = 0; kk < 32; ++kk) {
    int r0 = 4 * kk + 2 * hi;
    bb[kk].x = W01[r0 * NH + col];
    bb[kk].y = W01[(r0 + 1) * NH + col];
  }
  const float w2v = W2[col];

  for (int tile = wave_global; tile < n_tiles; tile += wave_stride) {
    const int base = tile * 16;

    // A operand: lane L holds x[row = base + L%16][4*kk + 2*(L/16) .. +1]
    int arow = base + col;
    if (arow >= n_atoms) arow = n_atoms - 1;  // clamp keeps EXEC all-ones
    const float* rowp = x + (size_t)arow * D_IN + 2 * hi;

    v8f c = {};
#pragma unroll
    for (int kk = 0; kk < 32; ++kk) {
      v2f a = __builtin_nontemporal_load((const v2f*)(rowp + 4 * kk));
      c = __builtin_amdgcn_wmma_f32_16x16x4_f32(
          /*neg_a=*/false, a, /*neg_b=*/false, bb[kk],
          /*c_mod=*/(short)0, c, /*reuse_a=*/false, /*reuse_b=*/false);
    }

    // pred[m] = sum_n h[m][n] * W2[n].
    // C/D layout: VGPR v, lanes0-15 -> (M=v, N=lane), lanes16-31 -> (M=v+8).
    float p = 0.0f;
#pragma unroll
    for (int v = 0; v < 8; ++v) {
      float t = c[v] * w2v;
      t += __shfl_xor(t, 1);   // masks <16 keep the two 16-lane halves separate
      t += __shfl_xor(t, 2);
      t += __shfl_xor(t, 4);
      t += __shfl_xor(t, 8);
      if ((lane & 7) == v) p = t;  // lanes 0-7: rows 0-7; lanes 16-23: rows 8-15
    }
    const bool active = ((lane & 15) < 8);
    if (!active) p = 0.0f;
    const int m = (lane & 7) + 8 * hi;
    const int a_idx = base + m;
    const bool valid = active && (a_idx < n_atoms);
    const int sid = structure[valid ? a_idx : 0];

    // Fast path: sorted ids -> whole tile usually maps to one structure.
    const int s0 = __shfl(sid, 0);
    const unsigned long long ACT = 0x00FF00FFull;  // lanes 0-7 and 16-23
    const unsigned long long eq = __ballot(sid == s0);
    if ((base + 16 <= n_atoms) && ((eq & ACT) == ACT)) {   // wave-uniform branch
      float t = p;
      t += __shfl_xor(t, 16);
      t += __shfl_xor(t, 8);
      t += __shfl_xor(t, 4);
      t += __shfl_xor(t, 2);
      t += __shfl_xor(t, 1);
      if (lane == 0) atomicAdd(&out[s0], t);
    } else {
      if (valid) atomicAdd(&out[sid], p);
    }
  }
}

// ---------------------------------------------------------------- launcher
extern "C" void kernel_launch(void* const* d_in, const int* in_sizes, int n_in,
                              void* d_out, int out_size, void* d_ws, size_t ws_size,
                              hipStream_t stream) {
  const float* x   = (const float*)d_in[0];
  const float* W0  = (const float*)d_in[1];
  const float* W1  = (const float*)d_in[2];
  const float* W2  = (const float*)d_in[3];
  const int* structure = (const int*)d_in[4];
  float* out = (float*)d_out;
  float* W01 = (float*)d_ws;  // 2048 floats of scratch

  const int n_atoms = in_sizes[0] / D_IN;
  const int n_tiles = (n_atoms + 15) / 16;

  zero_out_kernel<<<(out_size + 255) / 256, 256, 0, stream>>>(out, out_size);
  fuse_w01_kernel<<<(D_IN * NH + 255) / 256, 256, 0, stream>>>(W0, W1, W01);

  const int blocks = 2048;  // 8 waves/block -> 16384 waves, ~7.6 tiles each
  mlp_segsum_kernel<<<blocks, 256, 0, stream>>>(x, W01, W2, structure, out,
                                                n_atoms, n_tiles);
}